// FreqCounter_68315749810839
// MI455X (gfx1250) — compile-verified
//
#include <hip/hip_runtime.h>

// Pure gather: out[j] = lookup[item_ids[j]], 16.384M elements.
// Memory-bound: ~131 MB of streaming traffic + 480KB hot table.
// Strategy: NT 128-bit loads/stores for the two streams, RT gathers for the
// table (lives in L2/WGP$), and a one-shot global_prefetch_b8 sweep that pulls
// the whole 480KB table into cache before the gather storm.

typedef int   vi4 __attribute__((ext_vector_type(4)));
typedef float vf4 __attribute__((ext_vector_type(4)));

__global__ __launch_bounds__(256) void FreqCounter_gather_kernel(
    const int* __restrict__ item_ids,
    const float* __restrict__ lookup,
    float* __restrict__ out,
    int nvec,           // number of 4-element vector chunks
    int total,          // total elements
    int table_bytes)    // lookup table size in bytes
{
    const int tid = (int)blockIdx.x * 256 + (int)threadIdx.x;

    // Warm L2/WGP$ with the lookup table: first 15 blocks prefetch one 128B
    // cacheline per thread (15 * 256 * 128B = 480KB = whole table).
    // Lowers to global_prefetch_b8 on gfx1250.
    if (blockIdx.x < 15) {
        const int line = (int)blockIdx.x * 256 + (int)threadIdx.x;
        const int off  = line * 128;
        if (off < table_bytes) {
            __builtin_prefetch((const char*)lookup + off, /*rw=*/0, /*loc=*/3);
        }
    }

    if (tid < nvec) {
        // Streaming 128-bit NT load of 4 indices (don't pollute L2).
        const vi4 idx = __builtin_nontemporal_load((const vi4*)item_ids + tid);

        // Cached gathers from the hot 480KB table (regular temporal: stays
        // resident in L2 / WGP$ across the whole kernel).
        vf4 r;
        r.x = lookup[idx.x];
        r.y = lookup[idx.y];
        r.z = lookup[idx.z];
        r.w = lookup[idx.w];

        // Streaming 128-bit NT store (output is never re-read).
        __builtin_nontemporal_store(r, (vf4*)out + tid);
    } else {
        // Scalar tail (total % 4 elements), handled by threads past nvec.
        const int e = nvec * 4 + (tid - nvec);
        if (e < total) {
            out[e] = lookup[item_ids[e]];
        }
    }
}

extern "C" void kernel_launch(void* const* d_in, const int* in_sizes, int n_in,
                              void* d_out, int out_size, void* d_ws, size_t ws_size,
                              hipStream_t stream) {
    // setup_inputs order: user_ids (unused), item_ids, lookup
    const int*   item_ids = (const int*)d_in[1];
    const float* lookup   = (const float*)d_in[2];
    float*       out      = (float*)d_out;

    const int total       = out_size;            // 16,384,000
    const int nvec        = total / 4;           // 4,096,000 vec4 chunks
    const int rem         = total - nvec * 4;    // 0 for this shape
    const int nthreads    = nvec + rem;
    const int block       = 256;
    const int grid        = (nthreads + block - 1) / block;
    const int table_bytes = in_sizes[2] * 4;     // 120,000 floats = 480,000 B

    FreqCounter_gather_kernel<<<grid, block, 0, stream>>>(
        item_ids, lookup, out, nvec, total, table_bytes);
}